// SSIMLoss_13322988552270
// MI455X (gfx1250) — compile-verified
//
#include <hip/hip_runtime.h>

typedef float v2f __attribute__((ext_vector_type(2)));
typedef float v8f __attribute__((ext_vector_type(8)));

#define DIM   160
#define TS    16
#define HALO  5
#define IN_T  26          // TS + 2*HALO
#define NELE  (IN_T * IN_T)
#define SI_R  32          // padded rows (26 used)
#define SI_P  28          // padded cols (26 used; K-chunks read up to col 27)
#define P1_P  17          // pitch padding for bank-conflict-free 16-wide tiles
#define ZCH   40
#define NBLK  1600        // 4 batches * 100 xy-tiles * 4 z-chunks
#define C1V   (0.01f * 0.01f)
#define C2V   (0.03f * 0.03f)

// normalized 1-D Gaussian, ws=11, sigma=1.5 (matches reference)
__device__ __constant__ float c_g[11] = {
    0.001028380f, 0.007598758f, 0.036000776f, 0.109360687f, 0.213005542f,
    0.266011212f,
    0.213005542f, 0.109360687f, 0.036000776f, 0.007598758f, 0.001028380f};

__device__ __forceinline__ float bandw(int t) {
  return (t >= 0 && t <= 10) ? c_g[t] : 0.0f;
}

// Fused separable 3-D SSIM: x-conv and y-conv via V_WMMA_F32_16X16X4_F32,
// z-conv via per-thread sliding registers. One workgroup = 16x16 (x,y) tile
// streaming a 40-slice z chunk, with slice loads software-pipelined one
// iteration ahead of the WMMA compute.
__global__ void __launch_bounds__(256)
ssim3d_fused(const float* __restrict__ img1, const float* __restrict__ img2,
             double* __restrict__ gpart)
{
  __shared__ __align__(16) float sI1[SI_R][SI_P];
  __shared__ __align__(16) float sI2[SI_R][SI_P];
  __shared__ __align__(16) float sP1[5][SI_R][P1_P];
  __shared__ __align__(16) float sP2[5][TS][P1_P];
  __shared__ float sRed[256];

  const int tid    = threadIdx.x;
  const int lane   = tid & 31;
  // Force the wave id into an SGPR so all per-wave work partitioning compiles
  // to uniform scalar branches (EXEC stays all-1s around every WMMA).
  const int wave   = __builtin_amdgcn_readfirstlane(tid >> 5);
  const int laneLo = lane & 15;
  const int hi     = lane >> 4;

  // block decode: bid = ((n*100 + tile)*4 + zc)
  const int zc   = blockIdx.x & 3;
  const int tile = (blockIdx.x >> 2) % 100;
  const int n    = blockIdx.x / 400;
  const int tx = tile % 10, ty = tile / 10;
  const int x0 = tx * TS, y0 = ty * TS;
  const int zo0 = zc * ZCH;

  const float* __restrict__ b1 = img1 + (size_t)n * DIM * DIM * DIM;
  const float* __restrict__ b2 = img2 + (size_t)n * DIM * DIM * DIM;

  // Banded Gaussian operand. Identical per-lane values serve as the B-matrix
  // (4x16, pass-x) and the A-matrix (16x4, pass-y) per the CDNA5 WMMA f32
  // VGPR layouts: value pair = { G(4c + 2*hi - laneLo), G(4c + 2*hi+1 - laneLo) }.
  v2f wb[7];
#pragma unroll
  for (int c = 0; c < 7; ++c) {
    const int t0 = 4 * c + 2 * hi - laneLo;
    wb[c].x = bandw(t0);
    wb[c].y = bandw(t0 + 1);
  }

  // ---- phase-1 addressing, hoisted out of the z-loop (z-invariant) ----
  // element k of this thread: i = tid + 256k over the 26x26 slice
  int  ldsoff[3];   // LDS byte offset into sI*
  int  goff[3];     // global element offset within a z-plane
  bool gvalid[3];   // (element exists) && (gx,gy in range)
#pragma unroll
  for (int k = 0; k < 3; ++k) {
    const int i  = tid + k * 256;
    const int ii = (i < NELE) ? i : 0;
    const int yy = ii / IN_T;
    const int xx = ii - yy * IN_T;
    ldsoff[k] = (yy * SI_P + xx) * 4;
    const int gy = y0 + yy - HALO;
    const int gx = x0 + xx - HALO;
    gvalid[k] = (i < NELE) && ((unsigned)gy < (unsigned)DIM) &&
                ((unsigned)gx < (unsigned)DIM);
    goff[k] = gvalid[k] ? (gy * DIM + gx) : 0;
  }

  // zero LDS input tiles once; pad rows 26..31 and cols 26..27 stay zero forever
  for (int i = tid; i < SI_R * SI_P; i += 256) {
    (&sI1[0][0])[i] = 0.0f;
    (&sI2[0][0])[i] = 0.0f;
  }

  // sliding z-conv accumulators: acc[f][j] = partial for z_out = zi - 5 + j
  float acc[5][11];
#pragma unroll
  for (int f = 0; f < 5; ++f)
#pragma unroll
    for (int j = 0; j < 11; ++j) acc[f][j] = 0.0f;

  float ssum = 0.0f;
  const int px = tid & 15, py = tid >> 4;

  // hot-path z weights as immediates (compile-time constants)
  constexpr float GW[11] = {
      0.001028380f, 0.007598758f, 0.036000776f, 0.109360687f, 0.213005542f,
      0.266011212f,
      0.213005542f, 0.109360687f, 0.036000776f, 0.007598758f, 0.001028380f};

  float r1[3], r2[3];  // pipelined slice values (loaded one iteration ahead)

  auto issue_loads = [&](int zi) {
    const bool zok = (unsigned)zi < (unsigned)DIM;
    const int  zb  = zi * DIM * DIM;
#pragma unroll
    for (int k = 0; k < 3; ++k) {
      r1[k] = 0.0f;
      r2[k] = 0.0f;
      if (zok && gvalid[k]) {
        r1[k] = b1[zb + goff[k]];
        r2[k] = b2[zb + goff[k]];
      }
    }
  };

  auto commit_slice = [&]() {
    char* s1b = (char*)&sI1[0][0];
    char* s2b = (char*)&sI2[0][0];
#pragma unroll
    for (int k = 0; k < 3; ++k) {
      if (k < 2 || tid < (NELE - 512)) {
        *(float*)(s1b + ldsoff[k]) = r1[k];
        *(float*)(s2b + ldsoff[k]) = r2[k];
      }
    }
  };

  const int zfirst = zo0 - HALO;
  const int zlast  = zo0 + ZCH + HALO - 1;

  issue_loads(zfirst);   // prologue of the software pipeline
  __syncthreads();       // LDS zero-init visible

  for (int zi = zfirst; zi <= zlast; ++zi) {
    // ---- phase 1: commit pipelined slice, then prefetch the next one ----
    commit_slice();
    if (zi < zlast) issue_loads(zi + 1);  // completes during phases 2-4
    __syncthreads();

    // ---- phase 2: x-conv, 5 fields x 2 row-groups, WMMA f32 16x16x4 ----
    // out[y][x] = sum_k in[y][k] * G(k - x); 7 K-chunks of 4.
    // Units fully unrolled: f/rg are compile-time, guards are wave-uniform.
#pragma unroll
    for (int u = 0; u < 10; ++u) {
      if (wave == (u & 7)) {
        const int f  = u >> 1;   // constant after unroll
        const int rg = u & 1;    // constant after unroll
        const int row  = rg * 16 + laneLo;  // A: M = laneLo, K = 2*hi..2*hi+1
        const int colb = 2 * hi;
        v8f C = {0.f, 0.f, 0.f, 0.f, 0.f, 0.f, 0.f, 0.f};
#pragma unroll
        for (int c = 0; c < 7; ++c) {
          const int col = 4 * c + colb;
          const v2f a1 = *(const v2f*)&sI1[row][col];
          const v2f a2 = *(const v2f*)&sI2[row][col];
          v2f a;
          if      (f == 0) a = a1;
          else if (f == 1) a = a2;
          else if (f == 2) a = a1 * a1;
          else if (f == 3) a = a2 * a2;
          else             a = a1 * a2;
          C = __builtin_amdgcn_wmma_f32_16x16x4_f32(false, a, false, wb[c],
                                                    (short)0, C, false, false);
        }
#pragma unroll
        for (int r = 0; r < 8; ++r)
          sP1[f][rg * 16 + r + 8 * hi][laneLo] = C[r];
      }
    }
    __syncthreads();

    // ---- phase 3: y-conv, 5 fields, WMMA f32 16x16x4 ----
    // out[y][x] = sum_k G(k - y) * p1[k][x]; A = band weights, B = data.
#pragma unroll
    for (int f = 0; f < 5; ++f) {
      if (wave == f) {
        v8f C = {0.f, 0.f, 0.f, 0.f, 0.f, 0.f, 0.f, 0.f};
#pragma unroll
        for (int c = 0; c < 7; ++c) {
          const int k0 = 4 * c + 2 * hi;  // B: K = k0..k0+1, N = laneLo
          v2f bv;
          bv.x = sP1[f][k0][laneLo];
          bv.y = sP1[f][k0 + 1][laneLo];
          C = __builtin_amdgcn_wmma_f32_16x16x4_f32(false, wb[c], false, bv,
                                                    (short)0, C, false, false);
        }
#pragma unroll
        for (int r = 0; r < 8; ++r)
          sP2[f][r + 8 * hi][laneLo] = C[r];
      }
    }
    __syncthreads();

    // ---- phase 4: z accumulation (tap t = 10 - j), finalize z_out = zi-5 ----
    float v[5];
#pragma unroll
    for (int f = 0; f < 5; ++f) v[f] = sP2[f][py][px];
#pragma unroll
    for (int j = 0; j < 11; ++j) {
      const float w = GW[10 - j];
#pragma unroll
      for (int f = 0; f < 5; ++f)
        acc[f][j] = __builtin_fmaf(w, v[f], acc[f][j]);
    }
    const int zout = zi - HALO;
    if (zout >= zo0) {
      const float mu1 = acc[0][0], mu2 = acc[1][0];
      const float e11 = acc[2][0], e22 = acc[3][0], e12 = acc[4][0];
      const float mu1s = mu1 * mu1, mu2s = mu2 * mu2, mu12 = mu1 * mu2;
      const float s1 = e11 - mu1s, s2 = e22 - mu2s, s12 = e12 - mu12;
      const float num = (2.0f * mu12 + C1V) * (2.0f * s12 + C2V);
      const float den = (mu1s + mu2s + C1V) * (s1 + s2 + C2V);
      ssum += __fdividef(num, den);   // rcp+mul; ~2 ulp, fine at SSIM scale
    }
#pragma unroll
    for (int f = 0; f < 5; ++f) {
#pragma unroll
      for (int j = 0; j < 10; ++j) acc[f][j] = acc[f][j + 1];
      acc[f][10] = 0.0f;
    }
  }

  // ---- deterministic per-block partial: LDS tree reduce, write to gpart ----
  sRed[tid] = ssum;
  __syncthreads();
  for (int s = 128; s > 0; s >>= 1) {
    if (tid < s) sRed[tid] += sRed[tid + s];
    __syncthreads();
  }
  if (tid == 0) gpart[blockIdx.x] = (double)sRed[0];
}

// Deterministic final reduction over the 1600 block partials.
__global__ void __launch_bounds__(256)
ssim3d_final(const double* __restrict__ gpart, float* __restrict__ out)
{
  __shared__ double sd[256];
  double s = 0.0;
  for (int i = threadIdx.x; i < NBLK; i += 256) s += gpart[i];
  sd[threadIdx.x] = s;
  __syncthreads();
  for (int k = 128; k > 0; k >>= 1) {
    if (threadIdx.x < k) sd[threadIdx.x] += sd[threadIdx.x + k];
    __syncthreads();
  }
  if (threadIdx.x == 0)
    out[0] = 1.0f - (float)(sd[0] * (1.0 / 16384000.0));  // 4*160^3 voxels
}

extern "C" void kernel_launch(void* const* d_in, const int* in_sizes, int n_in,
                              void* d_out, int out_size, void* d_ws, size_t ws_size,
                              hipStream_t stream) {
  (void)in_sizes; (void)n_in; (void)out_size; (void)ws_size;
  const float* img1 = (const float*)d_in[0];
  const float* img2 = (const float*)d_in[1];
  float* out   = (float*)d_out;
  double* part = (double*)d_ws;  // 1600 doubles = 12.8 KB of scratch

  ssim3d_fused<<<dim3(NBLK), dim3(256), 0, stream>>>(img1, img2, part);
  ssim3d_final<<<dim3(1), dim3(256), 0, stream>>>(part, out);
}